// TILDE_Q_13546326852211
// MI455X (gfx1250) — compile-verified
//
#include <hip/hip_runtime.h>
#include <hip/hip_bf16.h>
#include <math.h>

#define B_DIM 2048
#define T_DIM 8192
#define NH    4096          // packed complex FFT length (T/2)
#define NBINS 4097          // rfft bins 0..4096
#define TOPK  5

typedef __attribute__((ext_vector_type(2))) float v2f;
typedef __attribute__((ext_vector_type(8))) float v8f;

#define PI_F 3.14159265358979323846f

// ---------------------------------------------------------------------------
// ws accumulator layout (floats): [0]=ashift  [1]=dom  [2]=nondom  [3]=amp
// ---------------------------------------------------------------------------
__global__ void tq_init_kernel(float* acc) {
    if (threadIdx.x < 4) acc[threadIdx.x] = 0.0f;
}

// ---------------------------------------------------------------------------
// ashift: per-row softmax(inp-tgt) over T, sum |1/T - p|, atomic into acc[0]
// ---------------------------------------------------------------------------
__global__ __launch_bounds__(256) void tq_ashift_kernel(const float* __restrict__ inp,
                                                        const float* __restrict__ tgt,
                                                        float* __restrict__ acc) {
    __shared__ float dbuf[T_DIM];   // 32 KB
    __shared__ float red[256];
    const int r = blockIdx.x, tid = threadIdx.x;
    const float* ip = inp + (size_t)r * T_DIM;
    const float* tp = tgt + (size_t)r * T_DIM;

    float mx = -3.402823466e38f;
    for (int t = tid; t < T_DIM; t += 256) {
        float d = ip[t] - tp[t];
        dbuf[t] = d;
        mx = fmaxf(mx, d);
    }
    red[tid] = mx; __syncthreads();
    for (int s = 128; s > 0; s >>= 1) { if (tid < s) red[tid] = fmaxf(red[tid], red[tid + s]); __syncthreads(); }
    mx = red[0]; __syncthreads();

    float se = 0.0f;
    for (int t = tid; t < T_DIM; t += 256) se += __expf(dbuf[t] - mx);
    red[tid] = se; __syncthreads();
    for (int s = 128; s > 0; s >>= 1) { if (tid < s) red[tid] += red[tid + s]; __syncthreads(); }
    const float invZ = 1.0f / red[0]; __syncthreads();

    const float u = 1.0f / (float)T_DIM;
    float s2 = 0.0f;
    for (int t = tid; t < T_DIM; t += 256) s2 += fabsf(u - __expf(dbuf[t] - mx) * invZ);
    red[tid] = s2; __syncthreads();
    for (int s = 128; s > 0; s >>= 1) { if (tid < s) red[tid] += red[tid + s]; __syncthreads(); }
    if (tid == 0) atomicAdd(acc + 0, red[0]);
}

// ---------------------------------------------------------------------------
// phase loss: packed-real 4096-pt in-place radix-2 FFT in LDS per row
// ---------------------------------------------------------------------------
__device__ __forceinline__ void tq_fft_inplace(float2* zbuf, int tid) {
    // bit-reversal permutation (12 bits), disjoint swap pairs
    for (int u1 = tid; u1 < NH; u1 += 256) {
        int rv = (int)(__brev((unsigned)u1) >> 20);
        if (rv > u1) {
            float2 a = zbuf[u1]; float2 b = zbuf[rv];
            zbuf[u1] = b; zbuf[rv] = a;
        }
    }
    __syncthreads();
    for (int s = 0; s < 12; s++) {
        const int half = 1 << s;
        for (int u1 = tid; u1 < NH / 2; u1 += 256) {
            const int j  = u1 & (half - 1);
            const int i0 = ((u1 >> s) << (s + 1)) + j;
            const int i1 = i0 + half;
            float sn, cs;
            __sincosf(-PI_F * (float)j / (float)half, &sn, &cs);
            float2 a = zbuf[i0], b = zbuf[i1];
            float2 t = make_float2(cs * b.x - sn * b.y, cs * b.y + sn * b.x);
            zbuf[i0] = make_float2(a.x + t.x, a.y + t.y);
            zbuf[i1] = make_float2(a.x - t.x, a.y - t.y);
        }
        __syncthreads();
    }
}

// reconstruct rfft bin k (0..4096) of the length-8192 real signal from Z
__device__ __forceinline__ float2 tq_unpackX(const float2* Z, int k) {
    const int i1 = k & (NH - 1);
    const int i2 = (NH - k) & (NH - 1);
    float2 zk = Z[i1];
    float2 zc = Z[i2];                       // conj below
    float er = 0.5f * (zk.x + zc.x);
    float ei = 0.5f * (zk.y - zc.y);
    float dr = zk.x - zc.x;
    float di = zk.y + zc.y;
    float orr = 0.5f * di;                   // O = (zk - conj(zc)) / (2i)
    float oi  = -0.5f * dr;
    float sn, cs;
    __sincosf(-2.0f * PI_F * (float)k / (float)T_DIM, &sn, &cs);
    return make_float2(er + (cs * orr - sn * oi), ei + (cs * oi + sn * orr));
}

__global__ __launch_bounds__(256) void tq_phase_kernel(const float* __restrict__ inp,
                                                       const float* __restrict__ tgt,
                                                       float* __restrict__ acc) {
    __shared__ float2 zbuf[NH];   // 32 KB
    __shared__ float  red[256];
    __shared__ int    redi[256];
    __shared__ int    topIdx[TOPK];
    __shared__ float2 tgtTop[TOPK];
    __shared__ float2 inpTop[TOPK];

    const int r = blockIdx.x, tid = threadIdx.x;
    const float2* tp2 = (const float2*)(tgt + (size_t)r * T_DIM);
    const float2* ip2 = (const float2*)(inp + (size_t)r * T_DIM);

    // ---- FFT(target row) ----
    for (int u = tid; u < NH; u += 256) zbuf[u] = tp2[u];
    __syncthreads();
    tq_fft_inplace(zbuf, tid);

    // ---- top-5 bins of |tgt_f| (bin 0 excluded), tie -> lower index ----
    for (int it = 0; it < TOPK; it++) {
        float bestV = -1.0f; int bestI = NBINS;
        for (int k = tid; k < NBINS; k += 256) {
            if (k == 0) continue;
            bool excl = false;
            for (int j = 0; j < it; j++) excl = excl || (topIdx[j] == k);
            if (excl) continue;
            float2 X = tq_unpackX(zbuf, k);
            float a2 = X.x * X.x + X.y * X.y;
            if (a2 > bestV) { bestV = a2; bestI = k; }
        }
        red[tid] = bestV; redi[tid] = bestI; __syncthreads();
        for (int s = 128; s > 0; s >>= 1) {
            if (tid < s) {
                if (red[tid + s] > red[tid] ||
                    (red[tid + s] == red[tid] && redi[tid + s] < redi[tid])) {
                    red[tid] = red[tid + s]; redi[tid] = redi[tid + s];
                }
            }
            __syncthreads();
        }
        if (tid == 0) topIdx[it] = redi[0];
        __syncthreads();
    }
    if (tid < TOPK) tgtTop[tid] = tq_unpackX(zbuf, topIdx[tid]);
    __syncthreads();

    // ---- FFT(input row) ----
    for (int u = tid; u < NH; u += 256) zbuf[u] = ip2[u];
    __syncthreads();
    tq_fft_inplace(zbuf, tid);

    // total spectral power of input (all rfft bins)
    float ssum = 0.0f;
    for (int k = tid; k < NBINS; k += 256) {
        float2 X = tq_unpackX(zbuf, k);
        ssum += X.x * X.x + X.y * X.y;
    }
    red[tid] = ssum; __syncthreads();
    for (int s = 128; s > 0; s >>= 1) { if (tid < s) red[tid] += red[tid + s]; __syncthreads(); }
    if (tid < TOPK) inpTop[tid] = tq_unpackX(zbuf, topIdx[tid]);
    __syncthreads();

    if (tid == 0) {
        float dom = 0.0f, topP = 0.0f;
        for (int j = 0; j < TOPK; j++) {
            float dx = tgtTop[j].x - inpTop[j].x;
            float dy = tgtTop[j].y - inpTop[j].y;
            dom  += dx * dx + dy * dy;
            topP += inpTop[j].x * inpTop[j].x + inpTop[j].y * inpTop[j].y;
        }
        atomicAdd(acc + 1, sqrtf(dom));
        atomicAdd(acc + 2, sqrtf(fmaxf(red[0] - topP, 0.0f)));
    }
}

// ---------------------------------------------------------------------------
// amp loss: C = tgt(2048x8192) * Bm^T, Bm[o,t] = tgt[o,T-1-t]-inp[o,T-1-t]
// 68.7 GFLOP fp32 GEMM via V_WMMA_F32_16X16X4_F32.
// Block tile 256(M)x128(N), 8 waves, 64x64 per wave (4x4 accumulators).
// LDS stores K-pairs contiguously (v2f) so each fragment is one ds_load_b64
// landing directly in an even-aligned VGPR pair (no repack movs).
// ---------------------------------------------------------------------------
__global__ __launch_bounds__(256) void tq_amp_gemm_kernel(const float* __restrict__ inp,
                                                          const float* __restrict__ tgt,
                                                          float* __restrict__ acc) {
    __shared__ v2f As2[256][9];   // [m][kpair] kpair=k/2, pad to 9 (stride 18 words)
    __shared__ v2f Bs2[128][9];   // [n][kpair]
    __shared__ float red[256];

    const int tid   = threadIdx.x;
    const int mTile = blockIdx.y * 256;
    const int nTile = blockIdx.x * 128;
    const int wave  = tid >> 5;
    const int lane  = tid & 31;
    const int wm    = (wave >> 1) * 64;          // wave M offset: 0/64/128/192
    const int wn    = (wave & 1) * 64;           // wave N offset: 0/64
    const int row   = lane & 15;                 // M (or N) within 16
    const int hl    = lane >> 4;                 // K-pair sub-offset: 0/1

    v8f acc8[4][4];
#pragma unroll
    for (int mi = 0; mi < 4; mi++)
#pragma unroll
        for (int ni = 0; ni < 4; ni++)
#pragma unroll
            for (int e = 0; e < 8; e++) acc8[mi][ni][e] = 0.0f;

    const float2* tgt2 = (const float2*)tgt;
    const float2* inp2 = (const float2*)inp;

    const int lmA = tid >> 3;                    // 0..31 (A stage rows per pass)
    const int lkp = tid & 7;                     // 0..7 k-pair

    for (int kb = 0; kb < T_DIM; kb += 16) {
        __syncthreads();
        // stage A tile: 256 rows x 8 k-pairs, coalesced float2 loads
#pragma unroll
        for (int i = 0; i < 8; i++) {
            const int m = lmA + i * 32;
            float2 va = tgt2[((size_t)(mTile + m) * T_DIM + kb) / 2 + lkp];
            v2f pa; pa.x = va.x; pa.y = va.y;
            As2[m][lkp] = pa;
        }
        // stage B tile: 128 rows x 8 k-pairs, flipped index, built on the fly
#pragma unroll
        for (int i = 0; i < 4; i++) {
            const int n = lmA + i * 32;
            const int ft1 = T_DIM - 2 - (kb + 2 * lkp);          // even
            const size_t g2 = ((size_t)(nTile + n) * T_DIM + ft1) / 2;
            float2 qt = tgt2[g2];
            float2 qi = inp2[g2];
            v2f pb; pb.x = qt.y - qi.y; pb.y = qt.x - qi.x;      // (Bm[k], Bm[k+1])
            Bs2[n][lkp] = pb;
        }
        if (kb + 16 < T_DIM) {
            __builtin_prefetch(&tgt[(size_t)(mTile + lmA) * T_DIM + (kb + 16 + 2 * lkp)], 0, 1);
        }
        __syncthreads();

#pragma unroll
        for (int kk = 0; kk < 4; kk++) {
            const int kp = kk * 2 + hl;          // k0 = 4*kk + 2*hl per WMMA A/B layout
            v2f a[4], b[4];
#pragma unroll
            for (int mi = 0; mi < 4; mi++) a[mi] = As2[wm + mi * 16 + row][kp];
#pragma unroll
            for (int ni = 0; ni < 4; ni++) b[ni] = Bs2[wn + ni * 16 + row][kp];
#pragma unroll
            for (int mi = 0; mi < 4; mi++)
#pragma unroll
                for (int ni = 0; ni < 4; ni++)
                    acc8[mi][ni] = __builtin_amdgcn_wmma_f32_16x16x4_f32(
                        false, a[mi], false, b[ni], (short)0, acc8[mi][ni], false, false);
        }
    }

    // |C| reduction: layout-agnostic (every (reg,lane) slot is one C element)
    float s = 0.0f;
#pragma unroll
    for (int mi = 0; mi < 4; mi++)
#pragma unroll
        for (int ni = 0; ni < 4; ni++)
#pragma unroll
            for (int e = 0; e < 8; e++) s += fabsf(acc8[mi][ni][e]);
    red[tid] = s; __syncthreads();
    for (int st = 128; st > 0; st >>= 1) { if (tid < st) red[tid] += red[tid + st]; __syncthreads(); }
    if (tid == 0) atomicAdd(acc + 3, red[0]);
}

// ---------------------------------------------------------------------------
__global__ void tq_combine_kernel(const float* __restrict__ acc, float* __restrict__ out) {
    if (threadIdx.x == 0 && blockIdx.x == 0) {
        const float invB = 1.0f / (float)B_DIM;
        float ashift = acc[0] * invB;
        float phase  = (acc[1] + acc[2]) * invB;
        float amp    = acc[3] * (invB * invB);
        out[0] = 0.5f * ashift + 0.5f * phase + 0.01f * amp;
    }
}

// ---------------------------------------------------------------------------
extern "C" void kernel_launch(void* const* d_in, const int* in_sizes, int n_in,
                              void* d_out, int out_size, void* d_ws, size_t ws_size,
                              hipStream_t stream) {
    (void)in_sizes; (void)n_in; (void)out_size; (void)ws_size;
    const float* inp = (const float*)d_in[0];
    const float* tgt = (const float*)d_in[1];
    float* out = (float*)d_out;
    float* acc = (float*)d_ws;

    tq_init_kernel<<<1, 32, 0, stream>>>(acc);
    tq_ashift_kernel<<<B_DIM, 256, 0, stream>>>(inp, tgt, acc);
    tq_phase_kernel<<<B_DIM, 256, 0, stream>>>(inp, tgt, acc);
    dim3 g(B_DIM / 128, B_DIM / 256);
    tq_amp_gemm_kernel<<<g, 256, 0, stream>>>(inp, tgt, acc);
    tq_combine_kernel<<<1, 32, 0, stream>>>(acc, out);
}